// Attn_2697239462679
// MI455X (gfx1250) — compile-verified
//
#include <hip/hip_runtime.h>
#include <math.h>

#define B_SZ 32
#define L_SZ 4096
#define H_SZ 1024

typedef __attribute__((ext_vector_type(2))) float v2f;
typedef __attribute__((ext_vector_type(4))) float v4f;
typedef __attribute__((ext_vector_type(8))) float v8f;

// ---------------------------------------------------------------------------
// Phase 1: v[b,n] = sum_k hidden[b,k] * W[k,n]   (32x1024 = 32x1024 @ 1024x1024)
// Full-precision f32 WMMA: V_WMMA_F32_16X16X4_F32, one 16x16 tile per wave.
// A 16x4 layout: lanes 0-15 hold M=0..15 with K={k,k+1}; lanes 16-31 K={k+2,k+3}.
// B 4x16 layout: N = lane&15, same K split. C/D: VGPR r -> M=r+8*(lane>=16), N=lane&15.
// ---------------------------------------------------------------------------
__global__ __launch_bounds__(256) void attn_hW_wmma(const float* __restrict__ hidden,
                                                    const float* __restrict__ W,
                                                    float* __restrict__ v)
{
    const int lane = threadIdx.x & 31;
    const int wg   = blockIdx.x * 8 + (threadIdx.x >> 5);   // global wave id 0..127
    const int mTile = wg >> 6;                              // 0..1   (B tiles)
    const int nTile = wg & 63;                              // 0..63  (H tiles)

    const int half = lane >> 4;       // 0: K pair {0,1}, 1: K pair {2,3}
    const int l15  = lane & 15;
    const int koff = half * 2;

    const float* aRow = hidden + (size_t)(mTile * 16 + l15) * H_SZ;   // A row for this lane
    const float* bCol = W + nTile * 16 + l15;                         // B column base

    v8f acc = {};
    for (int k = 0; k < H_SZ; k += 4) {
        v2f a, bm;
        a.x  = aRow[k + koff];
        a.y  = aRow[k + koff + 1];
        bm.x = bCol[(size_t)(k + koff) * H_SZ];
        bm.y = bCol[(size_t)(k + koff + 1) * H_SZ];
        acc = __builtin_amdgcn_wmma_f32_16x16x4_f32(
                /*neg_a=*/false, a, /*neg_b=*/false, bm,
                /*c_mod=*/(short)0, acc, /*reuse_a=*/false, /*reuse_b=*/false);
    }

    #pragma unroll
    for (int r = 0; r < 8; ++r) {
        const int mRow = mTile * 16 + r + half * 8;
        const int nCol = nTile * 16 + l15;
        v[(size_t)mRow * H_SZ + nCol] = acc[r];
    }
}

// ---------------------------------------------------------------------------
// Phase 1b: cbias[b] = sum_g bias[g] * hidden[b,g]. One wave per batch.
// ---------------------------------------------------------------------------
__global__ __launch_bounds__(1024) void attn_cbias(const float* __restrict__ hidden,
                                                   const float* __restrict__ bias,
                                                   float* __restrict__ cbias)
{
    const int lane = threadIdx.x & 31;
    const int b    = threadIdx.x >> 5;          // 0..31 (32 waves in the block)
    float acc = 0.f;
    for (int i = lane; i < H_SZ; i += 32)
        acc += hidden[(size_t)b * H_SZ + i] * bias[i];
    #pragma unroll
    for (int off = 16; off >= 1; off >>= 1)
        acc += __shfl_xor(acc, off, 32);
    if (lane == 0) cbias[b] = acc;
}

// ---------------------------------------------------------------------------
// Phase 2 (bandwidth-bound, ~512 MB read): e[b,l] = targets[b,l,:]·v_b + c_b.
// One wave per row; b128 coalesced loads; wave32 shuffle reduction.
// targets is a one-shot 512 MB stream (> 192 MB L2): load it NON-TEMPORAL so
// it doesn't thrash L2; v (128 KB, reused 131072x) then stays L2-resident.
// ---------------------------------------------------------------------------
__global__ __launch_bounds__(256) void attn_energy(const float* __restrict__ targets,
                                                   const float* __restrict__ v,
                                                   const float* __restrict__ cbias,
                                                   float* __restrict__ e)
{
    const int lane = threadIdx.x & 31;
    const long row = (long)blockIdx.x * 8 + (threadIdx.x >> 5);  // 0..B*L-1
    const int  b   = (int)(row >> 12);                            // row / 4096

    const v4f* t  = (const v4f*)(targets + (size_t)row * H_SZ);
    const v4f* vb = (const v4f*)(v + (size_t)b * H_SZ);

    float acc = 0.f;
    #pragma unroll
    for (int i = 0; i < 8; ++i) {
        const int idx = i * 32 + lane;          // 256 x 16B per row
        const v4f tv = __builtin_nontemporal_load(&t[idx]);   // stream: NT hint
        const v4f vv = vb[idx];                               // hot: keep RT (L2)
        acc += tv.x * vv.x + tv.y * vv.y + tv.z * vv.z + tv.w * vv.w;
    }
    #pragma unroll
    for (int off = 16; off >= 1; off >>= 1)
        acc += __shfl_xor(acc, off, 32);
    if (lane == 0) e[row] = acc + cbias[b];
}

// ---------------------------------------------------------------------------
// Phase 3: per-batch softmax over L=4096. One block per batch, 256 threads,
// 16 elements per thread kept in registers; LDS tree reductions for max & sum.
// ---------------------------------------------------------------------------
__global__ __launch_bounds__(256) void attn_softmax(const float* __restrict__ e,
                                                    float* __restrict__ out)
{
    __shared__ float red[256];
    const int b = blockIdx.x;
    const int t = threadIdx.x;

    float vals[16];
    float m = -INFINITY;
    #pragma unroll
    for (int i = 0; i < 16; ++i) {
        vals[i] = e[(size_t)b * L_SZ + t + i * 256];
        m = fmaxf(m, vals[i]);
    }
    red[t] = m; __syncthreads();
    for (int s = 128; s >= 1; s >>= 1) {
        if (t < s) red[t] = fmaxf(red[t], red[t + s]);
        __syncthreads();
    }
    m = red[0]; __syncthreads();

    float sum = 0.f;
    #pragma unroll
    for (int i = 0; i < 16; ++i) { vals[i] = __expf(vals[i] - m); sum += vals[i]; }
    red[t] = sum; __syncthreads();
    for (int s = 128; s >= 1; s >>= 1) {
        if (t < s) red[t] += red[t + s];
        __syncthreads();
    }
    const float inv = 1.f / red[0];
    #pragma unroll
    for (int i = 0; i < 16; ++i)
        out[(size_t)b * L_SZ + t + i * 256] = vals[i] * inv;
}

// ---------------------------------------------------------------------------
extern "C" void kernel_launch(void* const* d_in, const int* in_sizes, int n_in,
                              void* d_out, int out_size, void* d_ws, size_t ws_size,
                              hipStream_t stream)
{
    const float* hidden  = (const float*)d_in[0];   // [1,B,H]
    const float* targets = (const float*)d_in[1];   // [B,L,H]
    const float* W       = (const float*)d_in[2];   // [H,H]
    const float* bias    = (const float*)d_in[3];   // [H]
    float* out = (float*)d_out;                     // [B,1,L]

    float* ws = (float*)d_ws;
    float* v  = ws;                                 // B*H   floats (128 KB)
    float* cb = ws + (size_t)B_SZ * H_SZ;           // B     floats
    float* e  = cb + B_SZ;                          // B*L   floats (512 KB)

    attn_hW_wmma <<<16,            256, 0, stream>>>(hidden, W, v);
    attn_cbias   <<<1,             1024, 0, stream>>>(hidden, bias, cb);
    attn_energy  <<<(B_SZ*L_SZ)/8, 256, 0, stream>>>(targets, v, cb, e);
    attn_softmax <<<B_SZ,          256, 0, stream>>>(e, out);
}